// GCN_59846074302981
// MI455X (gfx1250) — compile-verified
//
#include <hip/hip_runtime.h>

typedef float v2f __attribute__((ext_vector_type(2)));
typedef float v4f __attribute__((ext_vector_type(4)));
typedef float v8f __attribute__((ext_vector_type(8)));

#define N_NODES 50000
#define N_EDGES 800000
#define FDIM 128
#define HDIM 128
#define CDIM 8
#define ROW_TILES 3125   // 50000 / 16

// ---------------- degree / norm ----------------
__global__ void k_init_deg(float* __restrict__ deg) {
    int i = blockIdx.x * blockDim.x + threadIdx.x;
    if (i < N_NODES) deg[i] = 1.0f;            // self-loop
}

__global__ void k_deg_scatter(const int* __restrict__ col, float* __restrict__ deg) {
    int e = blockIdx.x * blockDim.x + threadIdx.x;
    if (e < N_EDGES) atomicAdd(&deg[col[e]], 1.0f);
}

__global__ void k_dinv(float* __restrict__ deg) {
    int i = blockIdx.x * blockDim.x + threadIdx.x;
    if (i < N_NODES) deg[i] = rsqrtf(deg[i]);  // deg >= 1 always
}

// ---------------- GEMM: out[N x 128] = X[N x 128] @ W[128 x 128] ----------------
// One block = 16 rows x 128 cols; 8 waves, each wave owns one 16x16 col tile.
// W staged in LDS pre-swizzled: Wl[(k>>2)*512 + n*4 + (k&3)] = W[k][n]
__global__ void __launch_bounds__(256) k_gemm128(const float* __restrict__ X,
                                                 const float* __restrict__ W,
                                                 float* __restrict__ out) {
    __shared__ float Wl[FDIM * HDIM];          // 64 KB
    const int tid = threadIdx.x;
    for (int idx = tid; idx < FDIM * HDIM; idx += 256) {
        int k = idx >> 7, n = idx & 127;       // coalesced global read
        Wl[(k >> 2) * 512 + n * 4 + (k & 3)] = W[idx];
    }
    __syncthreads();

    const int wave = tid >> 5;
    const int lane = tid & 31;
    const int half = lane >> 4;                // 0: K0/K1, 1: K2/K3
    const int l16  = lane & 15;
    const int row0 = blockIdx.x * 16;
    const int col  = wave * 16 + l16;

    const float* Arow = X + (size_t)(row0 + l16) * FDIM + half * 2;
    const float* Bcol = Wl + col * 4 + half * 2;

    v8f acc = {};
#pragma unroll
    for (int k0 = 0; k0 < FDIM; k0 += 4) {
        v2f a = *(const v2f*)(Arow + k0);                  // X[m][k0+off .. +1]
        v2f b = *(const v2f*)(Bcol + (k0 >> 2) * 512);     // W[k0+off..][col]
        acc = __builtin_amdgcn_wmma_f32_16x16x4_f32(
            false, a, false, b, (short)0, acc, false, false);
    }
#pragma unroll
    for (int r = 0; r < 8; ++r)
        out[(size_t)(row0 + r + half * 8) * HDIM + col] = acc[r];
}

// ---------------- agg init: agg = bias + dinv^2 * h (self-loop term) ----------------
__global__ void __launch_bounds__(256) k_agg_init(const float* __restrict__ h,
                                                  const float* __restrict__ dinv,
                                                  const float* __restrict__ bias,
                                                  float* __restrict__ agg) {
    int t = blockIdx.x * blockDim.x + threadIdx.x;         // N*32 threads (float4 each)
    if (t >= N_NODES * 32) return;
    int i = t >> 5, j = t & 31;
    float d = dinv[i];
    float s = d * d;
    v4f hv = ((const v4f*)h)[(size_t)i * 32 + j];
    v4f bv = ((const v4f*)bias)[j];
    v4f o  = bv + s * hv;
    ((v4f*)agg)[(size_t)i * 32 + j] = o;
}

// ---------------- edge scatter: agg[col] += dinv[row]*dinv[col] * h[row] ----------------
// One wave per edge; each lane moves one float4 (32*4 = 128 features).
__global__ void __launch_bounds__(256) k_agg_edges(const int* __restrict__ row,
                                                   const int* __restrict__ col,
                                                   const float* __restrict__ dinv,
                                                   const float* __restrict__ h,
                                                   float* __restrict__ agg) {
    long gtid = (long)blockIdx.x * blockDim.x + threadIdx.x;
    int e = (int)(gtid >> 5);
    if (e >= N_EDGES) return;
    int lane = threadIdx.x & 31;
    int r = row[e];
    int c = col[e];
    float w = dinv[r] * dinv[c];
    v4f v = ((const v4f*)(h + (size_t)r * HDIM))[lane];
    float* dst = agg + (size_t)c * HDIM + lane * 4;
    atomicAdd(dst + 0, w * v.x);
    atomicAdd(dst + 1, w * v.y);
    atomicAdd(dst + 2, w * v.z);
    atomicAdd(dst + 3, w * v.w);
}

// ---------------- FC: out[N x 8] = X[N x 128] @ Wfc[128 x 8] + bfc ----------------
// Wfc padded to 16 cols of LDS zeros; one wave per 16-row tile, 8 tiles per block.
__global__ void __launch_bounds__(256) k_gemm_fc(const float* __restrict__ X,
                                                 const float* __restrict__ Wfc,
                                                 const float* __restrict__ bfc,
                                                 float* __restrict__ out) {
    __shared__ float Wl[FDIM * 16];            // 8 KB, swizzled like k_gemm128
    const int tid = threadIdx.x;
    for (int idx = tid; idx < FDIM * 16; idx += 256) {
        int k = idx >> 4, n = idx & 15;
        Wl[(k >> 2) * 64 + n * 4 + (k & 3)] = (n < CDIM) ? Wfc[k * CDIM + n] : 0.0f;
    }
    __syncthreads();

    const int wave = tid >> 5;
    const int lane = tid & 31;
    const int half = lane >> 4;
    const int l16  = lane & 15;
    const int tile = blockIdx.x * 8 + wave;
    if (tile >= ROW_TILES) return;             // wave-uniform guard, EXEC stays full
    const int row0 = tile * 16;

    const float* Arow = X + (size_t)(row0 + l16) * HDIM + half * 2;
    const float* Bcol = Wl + l16 * 4 + half * 2;

    v8f acc = {};
#pragma unroll
    for (int k0 = 0; k0 < HDIM; k0 += 4) {
        v2f a = *(const v2f*)(Arow + k0);
        v2f b = *(const v2f*)(Bcol + (k0 >> 2) * 64);
        acc = __builtin_amdgcn_wmma_f32_16x16x4_f32(
            false, a, false, b, (short)0, acc, false, false);
    }
    if (l16 < CDIM) {
        float bb = bfc[l16];
#pragma unroll
        for (int r = 0; r < 8; ++r)
            out[(size_t)(row0 + r + half * 8) * CDIM + l16] = acc[r] + bb;
    }
}

// ---------------- launcher ----------------
extern "C" void kernel_launch(void* const* d_in, const int* in_sizes, int n_in,
                              void* d_out, int out_size, void* d_ws, size_t ws_size,
                              hipStream_t stream) {
    const float* x   = (const float*)d_in[0];
    const int*   ei  = (const int*)  d_in[1];   // [2, E] flat: rows then cols
    const float* W1  = (const float*)d_in[2];
    const float* b1  = (const float*)d_in[3];
    const float* W2  = (const float*)d_in[4];
    const float* b2  = (const float*)d_in[5];
    const float* Wfc = (const float*)d_in[6];
    const float* bfc = (const float*)d_in[7];
    float* out = (float*)d_out;

    float* dinv = (float*)d_ws;                          // N floats (padded to 51200)
    float* h    = dinv + 51200;                          // N*H floats
    float* agg  = h + (size_t)N_NODES * HDIM;            // N*H floats

    const int* row = ei;
    const int* col = ei + N_EDGES;

    dim3 b256(256);
    // normalization
    k_init_deg   <<<(N_NODES + 255) / 256, b256, 0, stream>>>(dinv);
    k_deg_scatter<<<(N_EDGES + 255) / 256, b256, 0, stream>>>(col, dinv);
    k_dinv       <<<(N_NODES + 255) / 256, b256, 0, stream>>>(dinv);

    // layer 1
    k_gemm128  <<<ROW_TILES, b256, 0, stream>>>(x, W1, h);
    k_agg_init <<<(N_NODES * 32 + 255) / 256, b256, 0, stream>>>(h, dinv, b1, agg);
    k_agg_edges<<<(N_EDGES * 32 + 255) / 256, b256, 0, stream>>>(row, col, dinv, h, agg);

    // layer 2
    k_gemm128  <<<ROW_TILES, b256, 0, stream>>>(agg, W2, h);
    k_agg_init <<<(N_NODES * 32 + 255) / 256, b256, 0, stream>>>(h, dinv, b2, agg);
    k_agg_edges<<<(N_EDGES * 32 + 255) / 256, b256, 0, stream>>>(row, col, dinv, h, agg);

    // FC head
    k_gemm_fc<<<(ROW_TILES + 7) / 8, b256, 0, stream>>>(agg, Wfc, bfc, out);
}